// ESNNet_45054206935150
// MI455X (gfx1250) — compile-verified
//
#include <hip/hip_runtime.h>
#include <hip/hip_bf16.h>
#include <math.h>

// ---------------------------------------------------------------------------
// Problem constants (from reference): B=32, C=72, T=1000, F=64, H=512, NCLS=3
// ---------------------------------------------------------------------------
#define BB    32
#define CC    72
#define TT    1000
#define FF    64
#define HH    512
#define NCLS  3
#define KTOT  (FF + HH)          // 576 : concatenated K dim  [feats | h]
#define LEAK  0.1f
#define BN_EPS 1e-5f

// LDS layout (floats) for the scan kernel:
//   Xh   : 32 rows x HPADH   (h state, cols 0..511)
//   Xf[2]: 32 rows x FPAD    (feats double buffer, cols 0..63)
#define HPADH 514                // even -> float2-aligned rows
#define FPAD  66                 // 64 + 2 pad dwords (TDM pad_interval/amount)
#define XF0   (BB * HPADH)       // float offset of feats buffers
#define XFSZ  (BB * FPAD)        // floats per feats buffer
#define LDS_FLOATS (XF0 + 2 * XFSZ)

typedef __attribute__((ext_vector_type(2))) float v2f;
typedef __attribute__((ext_vector_type(8))) float v8f;
typedef __attribute__((ext_vector_type(4))) unsigned int v4u;
typedef __attribute__((ext_vector_type(8))) int v8i;
typedef __attribute__((ext_vector_type(4))) int v4i;

// ---------------------------------------------------------------------------
// Kernel 0: build Wcp, the concatenated weight [w_in^T ; W^T] (KTOT x HH),
// packed in (K/2, N, 2) order so each WMMA lane's (k,k+1) B-pair is one
// contiguous 8-byte load:  Wcp[((k>>1)*HH + n)*2 + (k&1)] = Wc[k][n].
// ---------------------------------------------------------------------------
__global__ __launch_bounds__(256) void build_wcp(const float* __restrict__ w_in,
                                                 const float* __restrict__ W,
                                                 float* __restrict__ Wcp) {
    int i = blockIdx.x * 256 + threadIdx.x;
    if (i >= KTOT * HH) return;
    int k = i >> 9;          // / 512
    int n = i & (HH - 1);    // % 512
    float v = (k < FF) ? w_in[n * FF + k] : W[n * HH + (k - FF)];
    Wcp[(((k >> 1) * HH + n) << 1) + (k & 1)] = v;
}

// ---------------------------------------------------------------------------
// Kernel 1: fused temporal-conv(1x5) + BN1 + ELU + spatial reduce over C +
//           BN2 + ELU  ->  feats (B, T, F) f32.  One thread per (b,t).
// ---------------------------------------------------------------------------
__device__ __forceinline__ float elu_f(float x) {
    return x > 0.0f ? x : expm1f(x);
}

__global__ __launch_bounds__(256) void feats_kernel(
    const float* __restrict__ x,        // (B,1,C,T)
    const float* __restrict__ w_temp,   // (F,1,1,5)
    const float* __restrict__ g1, const float* __restrict__ be1,
    const float* __restrict__ m1, const float* __restrict__ v1,
    const float* __restrict__ w_sp,     // (F,C)
    const float* __restrict__ g2, const float* __restrict__ be2,
    const float* __restrict__ m2, const float* __restrict__ v2,
    float* __restrict__ feats)          // (B,T,F)
{
    __shared__ float s_wt[FF * 5];
    __shared__ float s_ws[FF * CC];
    __shared__ float s_inv1[FF], s_off1[FF], s_inv2[FF], s_off2[FF];

    const int tid = threadIdx.x;
    for (int i = tid; i < FF * 5; i += 256)  s_wt[i] = w_temp[i];
    for (int i = tid; i < FF * CC; i += 256) s_ws[i] = w_sp[i];
    if (tid < FF) {
        float i1 = g1[tid] * rsqrtf(v1[tid] + BN_EPS);
        s_inv1[tid] = i1;
        s_off1[tid] = be1[tid] - m1[tid] * i1;
        float i2 = g2[tid] * rsqrtf(v2[tid] + BN_EPS);
        s_inv2[tid] = i2;
        s_off2[tid] = be2[tid] - m2[tid] * i2;
    }
    __syncthreads();

    const int gid = blockIdx.x * 256 + tid;       // 0 .. B*T-1 (exact)
    if (gid >= BB * TT) return;
    const int b = gid / TT;
    const int t = gid % TT;

    float acc[FF];
#pragma unroll
    for (int f = 0; f < FF; ++f) acc[f] = 0.0f;

    for (int c = 0; c < CC; ++c) {
        const float* xr = x + (b * CC + c) * TT;
        float xv[5];
#pragma unroll
        for (int k = 0; k < 5; ++k) {
            int tt = t - 2 + k;
            xv[k] = (tt >= 0 && tt < TT) ? xr[tt] : 0.0f;
        }
#pragma unroll
        for (int f = 0; f < FF; ++f) {
            const float* wt = &s_wt[f * 5];
            float conv = xv[0] * wt[0];
            conv = fmaf(xv[1], wt[1], conv);
            conv = fmaf(xv[2], wt[2], conv);
            conv = fmaf(xv[3], wt[3], conv);
            conv = fmaf(xv[4], wt[4], conv);
            float e = elu_f(fmaf(conv, s_inv1[f], s_off1[f]));
            acc[f] = fmaf(s_ws[f * CC + c], e, acc[f]);
        }
    }

    float* out = feats + (b * TT + t) * FF;
#pragma unroll
    for (int f = 0; f < FF; ++f) {
        out[f] = elu_f(fmaf(acc[f], s_inv2[f], s_off2[f]));
    }
}

// ---------------------------------------------------------------------------
// Kernel 2: ESN scan, single workgroup (32 wave32 waves).
//   Z(32x512) = [feats_t | h] (32x576) @ Wc (576x512) via f32 WMMA 16x16x4.
//   Wave w owns N-tile w, M-tiles 0 and 1; K-loop split into the feats part
//   (double-buffered LDS slab, filled by the Tensor Data Mover for t+1 while
//   step t computes) and the h part (fixed LDS slab).
//   h_new = 0.9*h + 0.1*tanh(z + b_in); mean-pool kept in VGPRs; classifier
//   (pooled @ w_cls^T + b_cls) at the end of the same kernel.
// ---------------------------------------------------------------------------
__global__ __launch_bounds__(1024) void esn_scan(
    const float* __restrict__ feats,   // (B,T,F)
    const float* __restrict__ Wcp,     // (KTOT/2, HH, 2) paired K-major
    const float* __restrict__ b_in,    // (H)
    const float* __restrict__ w_cls,   // (NCLS, H)
    const float* __restrict__ b_cls,   // (NCLS)
    float* __restrict__ out)           // (B, NCLS)
{
    extern __shared__ float smem[];          // LDS_FLOATS floats
    float* Xh = smem;                        // 32 x HPADH
    float* Xf = smem + XF0;                  // 2 x (32 x FPAD)

    const int tid  = threadIdx.x;
    const int lane = tid & 31;
    const int wv   = tid >> 5;               // wave id 0..31 == N-tile
    const int half = lane >> 4;              // 0/1 (K sub-pair select)
    const int l15  = lane & 15;

    // ---- init: h = 0, feats t=0 into buffer 0 -----------------------------
    for (int i = tid; i < XF0; i += 1024) Xh[i] = 0.0f;
    for (int i = tid; i < BB * FF; i += 1024) {
        int b = i >> 6, f = i & (FF - 1);
        Xf[b * FPAD + f] = feats[(b * TT) * FF + f];
    }
    __syncthreads();

    const int   nb  = (wv << 4) + l15;       // lane's N column
    const float bin = b_in[nb];

    // TDM D# group-1 constants: data_size=4B, pad_enable, pad_interval=64dw,
    // pad_amount=2dw, tensor 64x32 (dim0 x dim1), dim0_stride = T*F = 64000.
    const int g1w0 = 0x03520000;             // pad_amt=1|pad_int=5|pad_en|ds=2
    const unsigned int ldsBase = (unsigned int)(uintptr_t)smem;

    v8f sum0 = {}; v8f sum1 = {};
    const int fr0 = l15 * FPAD,  fr1 = (16 + l15) * FPAD;
    const int hr0 = l15 * HPADH, hr1 = (16 + l15) * HPADH;
    const int koff = half << 1;              // lanes 16-31 take K+2,K+3

    int parity = 0;
    for (int t = 0; t < TT; ++t) {
        // ---- kick off async TDM fill of the other feats buffer (t+1) -----
        if (wv == 0 && t + 1 < TT) {
            unsigned long long ga =
                (unsigned long long)(uintptr_t)(feats + (size_t)(t + 1) * FF);
            v4u g0;
            g0.x = 1u;                                        // count = 1
            g0.y = ldsBase + (unsigned int)((XF0 + (parity ^ 1) * XFSZ) * 4);
            g0.z = (unsigned int)ga;                          // gaddr[31:0]
            g0.w = (unsigned int)((ga >> 32) & 0x1FFFFFFu) | (2u << 30);
            v8i g1;
            g1[0] = g1w0;
            g1[1] = FF << 16;          // tensor_dim0[15:0] = 64
            g1[2] = BB << 16;          // tensor_dim1[15:0] = 32
            g1[3] = FF << 16;          // tile_dim0 = 64
            g1[4] = BB;                // tile_dim1 = 32
            g1[5] = TT * FF;           // tensor_dim0_stride = 64000
            g1[6] = 0;
            g1[7] = 0;
            v4i gz4 = {0, 0, 0, 0};
            v8i gz8 = {0, 0, 0, 0, 0, 0, 0, 0};
            __builtin_amdgcn_tensor_load_to_lds(g0, g1, gz4, gz4, gz8, 0);
        }

        // ---- K loop: feats part (K = 0..63) ------------------------------
        v8f acc0 = {}; v8f acc1 = {};
        const float* fb = &Xf[parity * XFSZ];
#pragma unroll 4
        for (int kk = 0; kk < FF / 4; ++kk) {
            const int k = (kk << 2) + koff;
            v2f a0 = *(const v2f*)&fb[fr0 + k];
            v2f a1 = *(const v2f*)&fb[fr1 + k];
            v2f bv = *(const v2f*)&Wcp[(((2 * kk + half) * HH) + nb) << 1];
            acc0 = __builtin_amdgcn_wmma_f32_16x16x4_f32(
                       false, a0, false, bv, (short)0, acc0, false, false);
            acc1 = __builtin_amdgcn_wmma_f32_16x16x4_f32(
                       false, a1, false, bv, (short)0, acc1, false, false);
        }
        // ---- K loop: h part (K = 64..575) --------------------------------
#pragma unroll 4
        for (int kk = 0; kk < HH / 4; ++kk) {
            const int j = (kk << 2) + koff;
            v2f a0 = *(const v2f*)&Xh[hr0 + j];
            v2f a1 = *(const v2f*)&Xh[hr1 + j];
            v2f bv = *(const v2f*)&Wcp[(((32 + 2 * kk + half) * HH) + nb) << 1];
            acc0 = __builtin_amdgcn_wmma_f32_16x16x4_f32(
                       false, a0, false, bv, (short)0, acc0, false, false);
            acc1 = __builtin_amdgcn_wmma_f32_16x16x4_f32(
                       false, a1, false, bv, (short)0, acc1, false, false);
        }
        __syncthreads();   // everyone done reading Xh/Xf[parity]

        // ---- leaky-tanh update; each lane owns exactly its (m, nb) slots -
#pragma unroll
        for (int r = 0; r < 8; ++r) {
            int m0 = r + (half << 3);
            int m1 = m0 + 16;
            float h0 = Xh[m0 * HPADH + nb];
            float h1 = Xh[m1 * HPADH + nb];
            float hn0 = fmaf(1.0f - LEAK, h0, LEAK * tanhf(acc0[r] + bin));
            float hn1 = fmaf(1.0f - LEAK, h1, LEAK * tanhf(acc1[r] + bin));
            Xh[m0 * HPADH + nb] = hn0;
            Xh[m1 * HPADH + nb] = hn1;
            sum0[r] += hn0;
            sum1[r] += hn1;
        }

        // TDM fill of Xf[parity^1] must land before anyone reads it next step
        __builtin_amdgcn_s_wait_tensorcnt(0);   // no-op for non-issuing waves
        __syncthreads();
        parity ^= 1;
    }

    // ---- pooled = mean over T, staged into the h slab ---------------------
    const float invT = 1.0f / (float)TT;
#pragma unroll
    for (int r = 0; r < 8; ++r) {
        int m0 = r + (half << 3);
        int m1 = m0 + 16;
        Xh[m0 * HPADH + nb] = sum0[r] * invT;
        Xh[m1 * HPADH + nb] = sum1[r] * invT;
    }
    __syncthreads();

    // ---- classifier: (B x H) @ (H x NCLS) ---------------------------------
    if (tid < BB * NCLS) {
        int b = tid / NCLS, c = tid % NCLS;
        float acc = b_cls[c];
        for (int j = 0; j < HH; ++j)
            acc = fmaf(Xh[b * HPADH + j], w_cls[c * HH + j], acc);
        out[b * NCLS + c] = acc;
    }
}

// ---------------------------------------------------------------------------
// Host-side launcher
// ---------------------------------------------------------------------------
extern "C" void kernel_launch(void* const* d_in, const int* in_sizes, int n_in,
                              void* d_out, int out_size, void* d_ws, size_t ws_size,
                              hipStream_t stream) {
    const float* x      = (const float*)d_in[0];
    const float* w_temp = (const float*)d_in[1];
    const float* g1     = (const float*)d_in[2];
    const float* be1    = (const float*)d_in[3];
    const float* m1     = (const float*)d_in[4];
    const float* v1     = (const float*)d_in[5];
    const float* w_sp   = (const float*)d_in[6];
    const float* g2     = (const float*)d_in[7];
    const float* be2    = (const float*)d_in[8];
    const float* m2     = (const float*)d_in[9];
    const float* v2     = (const float*)d_in[10];
    const float* w_in   = (const float*)d_in[11];
    const float* b_in   = (const float*)d_in[12];
    const float* W      = (const float*)d_in[13];
    const float* w_cls  = (const float*)d_in[14];
    const float* b_cls  = (const float*)d_in[15];
    float* out = (float*)d_out;

    // workspace layout (floats): Wcp [KTOT*HH] | feats [B*T*F]
    float* Wcp   = (float*)d_ws;
    float* feats = Wcp + (size_t)KTOT * HH;

    {   // concat + pair-pack [w_in^T ; W^T] -> Wcp
        int n = KTOT * HH;
        build_wcp<<<(n + 255) / 256, 256, 0, stream>>>(w_in, W, Wcp);
    }
    {   // fused frontend -> feats
        int n = BB * TT;
        feats_kernel<<<(n + 255) / 256, 256, 0, stream>>>(
            x, w_temp, g1, be1, m1, v1, w_sp, g2, be2, m2, v2, feats);
    }
    {   // single-WGP WMMA scan (TDM-prefetched feats) + pool + classifier
        size_t lds_bytes = (size_t)LDS_FLOATS * sizeof(float);  // ~82.7 KB
        esn_scan<<<1, 1024, lds_bytes, stream>>>(feats, Wcp, b_in, w_cls,
                                                 b_cls, out);
    }
}